// bce_wlsep_50483045597975
// MI455X (gfx1250) — compile-verified
//
#include <hip/hip_runtime.h>
#include <math.h>

// ---------------------------------------------------------------------------
// bce + wlsep fused loss for MI455X (gfx1250, wave32).
//
// Algebraic collapse of the O(B*C^2) reference to O(B*C):
//   L_b   = logsumexp_{i: lab=0}(x_i)         (per row)
//   lse_j = softplus(L_b - x_j)  for positives j   (exp(x_i-x_j)=exp(x_i)exp(-x_j))
//   wlsep = sum lse_j / #pos ;  bce = mean(softplus(lab ? -x : x))
//
// -> ~2 MB input, one global pass, VALU/latency bound. One wave per 16-row
// tile (256 workgroups) for full-chip spread. The row exp-sum uses
// V_WMMA_F32_16X16X4_F32 as D = A x ones + C (full fp32 precision).
// Native v_exp/v_log keep everything branchless so EXEC stays all-1s.
// ---------------------------------------------------------------------------

typedef __attribute__((ext_vector_type(2))) float v2f;
typedef __attribute__((ext_vector_type(8))) float v8f;

#define C_DIM 128
#define ROWS_PER_WAVE 16
#define NPAIR 32            // 32 float2 pairs = 64 elements per lane

__device__ __forceinline__ float softplus_fast(float t) {
    // log(1 + exp(t)) with native transcendentals; exact 0 for t = -inf.
    float e = __expf(-fabsf(t));
    return fmaxf(t, 0.0f) + __logf(1.0f + e);
}

__device__ __forceinline__ float wred_sum(float v) {
    #pragma unroll
    for (int off = 16; off > 0; off >>= 1)
        v += __shfl_xor(v, off, 32);
    return v;
}

__global__ void __launch_bounds__(32)
bce_wlsep_rows_kernel(const float* __restrict__ X,
                      const int*   __restrict__ T,
                      float* __restrict__ partials) {   // [gridDim.x * 3]
    const int lane = threadIdx.x & 31;
    const int m    = lane & 15;         // row within this wave's 16-row tile
    const int half = lane >> 4;         // selects cols {0,1} vs {2,3} mod 4 (WMMA A K-layout)
    const int row  = blockIdx.x * ROWS_PER_WAVE + m;

    const float* __restrict__ xrow = X + (size_t)row * C_DIM;
    const int*   __restrict__ trow = T + (size_t)row * C_DIM;
    const float2* __restrict__ xp = reinterpret_cast<const float2*>(xrow);
    const int2*   __restrict__ tp = reinterpret_cast<const int2*>(trow);

    // ---- single global pass: 64 elements/lane into registers + label bitmasks
    float xs0[NPAIR], xs1[NPAIR];
    unsigned msk0 = 0u, msk1 = 0u;      // bit j = label of col 4j+2*half (+1)
    #pragma unroll
    for (int j = 0; j < NPAIR; ++j) {
        const int p = (4 * j + 2 * half) >> 1;   // float2 index, 8B aligned
        float2 xv = xp[p];
        int2   lv = tp[p];
        xs0[j] = xv.x;  xs1[j] = xv.y;
        msk0 |= (lv.x != 0 ? 1u : 0u) << j;
        msk1 |= (lv.y != 0 ? 1u : 0u) << j;
    }

    // ---- pass 1 (registers): bce partial, positive count, neg-row-max ------
    float bce = 0.0f;
    float negmax = -INFINITY;
    #pragma unroll
    for (int j = 0; j < NPAIR; ++j) {
        const bool b0 = (msk0 >> j) & 1u;
        const bool b1 = (msk1 >> j) & 1u;
        const float x0 = xs0[j], x1 = xs1[j];
        bce += softplus_fast(b0 ? -x0 : x0);
        bce += softplus_fast(b1 ? -x1 : x1);
        negmax = fmaxf(negmax, b0 ? -INFINITY : x0);
        negmax = fmaxf(negmax, b1 ? -INFINITY : x1);
    }
    float pos = (float)(__popc(msk0) + __popc(msk1));
    // lanes L and L^16 hold complementary halves of row m
    float rowmax = fmaxf(negmax, __shfl_xor(negmax, 16, 32));

    // ---- pass 2: S_b = sum_{neg i} exp(x_i - rowmax) via f32 WMMA rowsum ---
    // A (16x4 f32) layout: lanes 0-15 hold K=0,1 ; lanes 16-31 hold K=2,3.
    // B = all-ones (4x16) -> layout-agnostic. Two accumulators break the
    // serial C-chain; summed at the end.
    v8f accA = {0.f,0.f,0.f,0.f,0.f,0.f,0.f,0.f};
    v8f accB = {0.f,0.f,0.f,0.f,0.f,0.f,0.f,0.f};
    const v2f bones = {1.0f, 1.0f};
    #pragma unroll
    for (int j = 0; j < NPAIR; j += 2) {
        v2f a0, a1;
        a0[0] = ((msk0 >> j) & 1u)     ? 0.0f : __expf(xs0[j]     - rowmax);
        a0[1] = ((msk1 >> j) & 1u)     ? 0.0f : __expf(xs1[j]     - rowmax);
        a1[0] = ((msk0 >> (j+1)) & 1u) ? 0.0f : __expf(xs0[j + 1] - rowmax);
        a1[1] = ((msk1 >> (j+1)) & 1u) ? 0.0f : __expf(xs1[j + 1] - rowmax);
        accA = __builtin_amdgcn_wmma_f32_16x16x4_f32(false, a0, false, bones,
                                                     (short)0, accA, false, false);
        accB = __builtin_amdgcn_wmma_f32_16x16x4_f32(false, a1, false, bones,
                                                     (short)0, accB, false, false);
    }
    v8f acc = accA + accB;

    // D layout: lanes 0-15 hold rows 0-7 in acc[0..7]; lanes 16-31 rows 8-15.
    // LDS bounce so each lane picks up S for its own row m.
    __shared__ float Srow[ROWS_PER_WAVE];
    if (lane == 0) {
        #pragma unroll
        for (int v = 0; v < 8; ++v) Srow[v] = acc[v];        // rows 0..7  (N=0)
    }
    if (lane == 16) {
        #pragma unroll
        for (int v = 0; v < 8; ++v) Srow[8 + v] = acc[v];    // rows 8..15 (N=0)
    }
    __syncthreads();    // single-wave workgroup: lowers to S_NOP, DS order kept
    const float S = Srow[m];

    // ---- pass 3 (registers): sum softplus(L_b - x_j) over positives --------
    // Branchless even when the row has no negatives: S=0, rowmax=-inf
    // -> Lb=-inf -> softplus(-inf)=0, matching log(exp(0)) = 0.
    const float Lb = rowmax + __logf(S);
    float wl = 0.0f;
    #pragma unroll
    for (int j = 0; j < NPAIR; ++j) {
        const bool b0 = (msk0 >> j) & 1u;
        const bool b1 = (msk1 >> j) & 1u;
        wl += b0 ? softplus_fast(Lb - xs0[j]) : 0.0f;
        wl += b1 ? softplus_fast(Lb - xs1[j]) : 0.0f;
    }

    // ---- wave reduction, one partial triple per workgroup (deterministic) --
    bce = wred_sum(bce);
    wl  = wred_sum(wl);
    pos = wred_sum(pos);
    if (lane == 0) {
        partials[blockIdx.x * 3 + 0] = bce;
        partials[blockIdx.x * 3 + 1] = wl;
        partials[blockIdx.x * 3 + 2] = pos;
    }
}

__global__ void __launch_bounds__(32)
bce_wlsep_finalize_kernel(const float* __restrict__ partials,
                          float* __restrict__ out,
                          int nblocks, float inv_total) {
    const int lane = threadIdx.x;
    float sb = 0.f, sw = 0.f, sp = 0.f;
    for (int i = lane; i < nblocks; i += 32) {
        sb += partials[3 * i + 0];
        sw += partials[3 * i + 1];
        sp += partials[3 * i + 2];
    }
    #pragma unroll
    for (int off = 16; off > 0; off >>= 1) {
        sb += __shfl_xor(sb, off, 32);
        sw += __shfl_xor(sw, off, 32);
        sp += __shfl_xor(sp, off, 32);
    }
    if (lane == 0) {
        const float bce_loss   = sb * inv_total;
        const float wlsep_loss = (sp > 0.f) ? (sw / sp) : 0.f;
        out[0] = bce_loss + wlsep_loss;
    }
}

extern "C" void kernel_launch(void* const* d_in, const int* in_sizes, int n_in,
                              void* d_out, int out_size, void* d_ws, size_t ws_size,
                              hipStream_t stream) {
    (void)n_in; (void)out_size; (void)ws_size;
    const float* X = (const float*)d_in[0];   // output: [B, 128] float32 logits
    const int*   T = (const int*)d_in[1];     // target: [B, 128] int32 {0,1}
    float* out = (float*)d_out;
    float* partials = (float*)d_ws;

    const int Brows   = in_sizes[0] / C_DIM;           // 4096
    const int nblocks = Brows / ROWS_PER_WAVE;         // 256 single-wave WGs
    const float inv_total = 1.0f / (float)(Brows * C_DIM);

    bce_wlsep_rows_kernel<<<nblocks, 32, 0, stream>>>(X, T, partials);
    bce_wlsep_finalize_kernel<<<1, 32, 0, stream>>>(partials, out, nblocks, inv_total);
}